// GCN_IMGSNP_75007308858121
// MI455X (gfx1250) — compile-verified
//
#include <hip/hip_runtime.h>
#include <hip/hip_bf16.h>

// ---------------------------------------------------------------------------
// GCN_IMGSNP on MI455X (gfx1250, wave32).
// - per-graph dense adjacency in LDS turns edge gather/scatter into f32 WMMA
// - head GEMMs: WMMA with double-buffered GLOBAL_LOAD_ASYNC_TO_LDS_B128 A-tile
//   staging (ASYNCcnt), weights resident in 192MB L2.
// ---------------------------------------------------------------------------

typedef float v2f __attribute__((ext_vector_type(2)));
typedef float v8f __attribute__((ext_vector_type(8)));

#define ROIS   90
#define BGR    1024
#define NNODES (BGR * ROIS)      // 92160
#define EPG    4096
#define NEDGE  (BGR * EPG)       // 4194304
#define HID    64
#define CATW   (3 * HID)         // 192
#define LIN1K  (ROIS * CATW)     // 17280
#define FEATP  288               // 270 padded to /4 and /32
#define FEATV  (ROIS * 3)        // 270

// async DMA: global memory -> LDS, 16 bytes, tracked by ASYNCcnt.
// lds_off = low 32 bits of generic shared pointer (= LDS byte address).
__device__ __forceinline__ void async_ld_b128(unsigned lds_off, const float* gaddr) {
    asm volatile("global_load_async_to_lds_b128 %0, %1, off"
                 :: "v"(lds_off), "v"((unsigned long long)(uintptr_t)gaddr)
                 : "memory");
}
__device__ __forceinline__ void wait_async_le2() {
    asm volatile("s_wait_asynccnt 0x2" ::: "memory");
}
__device__ __forceinline__ void wait_async_le0() {
    asm volatile("s_wait_asynccnt 0x0" ::: "memory");
}

// ---------------------------------------------------------------- utilities
__global__ void k_zero(float* __restrict__ p, int n) {
    int i = blockIdx.x * blockDim.x + threadIdx.x;
    if (i < n) p[i] = 0.0f;
}

__global__ void k_deg(const int* __restrict__ edst,
                      const float* __restrict__ ew,
                      float* __restrict__ deg) {
    int e = blockIdx.x * blockDim.x + threadIdx.x;
    if (e < NEDGE) atomicAdd(&deg[edst[e]], ew[e]);
}

__global__ void k_dinv(const float* __restrict__ deg, float* __restrict__ dinv) {
    int i = blockIdx.x * blockDim.x + threadIdx.x;
    if (i < NNODES) dinv[i] = rsqrtf(deg[i] + 1.0f);
}

__global__ void k_gate(const float* __restrict__ x,
                       const float* __restrict__ prob,
                       float* __restrict__ img_feat) {
    int i = blockIdx.x * blockDim.x + threadIdx.x;  // over 1024*288
    if (i >= BGR * FEATP) return;
    int b = i / FEATP, col = i % FEATP;
    float v = 0.0f;
    if (col < FEATV) {
        int roi = col / 3, f = col - roi * 3;
        v = x[(size_t)(b * ROIS + roi) * 3 + f] * prob[roi * 3 + f];
    }
    img_feat[i] = v;
}

// ------------------------------------------------------------ GCN layer
// One block per graph. Phases (LDS reused, 64 KB total):
//  1) load W (F_INx64) + H (90xF_IN) -> compute HW[96][64]
//  2) zero A[96][96], load dinv[90]; edge loop: A[d][s] += norm (ds_add_f32)
//     diag: A[i][i] += dinv[i]^2
//  3) OUT = A @ HW via v_wmma_f32_16x16x4_f32; +bias, ReLU, store into
//     img_out cat layout: img_out[g*17280 + roi*192 + out_off + f]
template <int F_IN>
__global__ __launch_bounds__(256)
void k_gcn(const float* __restrict__ in, int g_stride, int row_stride, int col_off,
           const float* __restrict__ W, const float* __restrict__ bias,
           const int* __restrict__ esrc, const int* __restrict__ edst,
           const float* __restrict__ ew, const float* __restrict__ dinv,
           float* __restrict__ img_out, int out_off)
{
    __shared__ float smem[16384];               // 64 KB
    float* sHW  = smem;                          // [96][64]   (all phases)
    float* sW   = smem + 6144;                   // [F_IN][64] (phase 1)
    float* sH   = smem + 6144 + F_IN * 64;       // [96][F_IN] (phase 1)
    float* sA   = smem + 6144;                   // [96][96]   (phase 2/3)
    float* sDnv = smem + 15360;                  // [96]       (phase 2/3)

    const int g    = blockIdx.x;
    const int tid  = threadIdx.x;
    const int lane = tid & 31;
    const int wave = tid >> 5;
    const int l15  = lane & 15;
    const int hi   = lane >> 4;                  // 0: lanes 0-15, 1: lanes 16-31

    // ---- phase 1: HW = H @ W ------------------------------------------------
    if constexpr (F_IN == 64) {
        // 16B-aligned rows: vectorized B128 loads
        for (int i = tid; i < 64 * 16; i += 256)       // W: 64x64
            *(float4*)&sW[i * 4] = *(const float4*)&W[i * 4];
        for (int i = tid; i < 96 * 16; i += 256) {     // H: 96x64 (pad rows 0)
            int r = i >> 4, q = i & 15;
            float4 v = make_float4(0.f, 0.f, 0.f, 0.f);
            if (r < ROIS)
                v = *(const float4*)&in[(size_t)g * g_stride + r * row_stride +
                                        col_off + q * 4];
            *(float4*)&sH[r * 64 + q * 4] = v;
        }
    } else {
        for (int i = tid; i < F_IN * 64; i += 256) sW[i] = W[i];
        for (int i = tid; i < 96 * F_IN; i += 256) {
            int r = i / F_IN, c = i - r * F_IN;
            sH[i] = (r < ROIS)
                  ? in[(size_t)g * g_stride + r * row_stride + col_off + c]
                  : 0.0f;
        }
    }
    __syncthreads();
    for (int i = tid; i < 96 * 64; i += 256) {
        int r = i >> 6, c = i & 63;
        float acc = 0.0f;
#pragma unroll
        for (int k = 0; k < F_IN; ++k)
            acc = fmaf(sH[r * F_IN + k], sW[k * 64 + c], acc);
        sHW[i] = acc;                            // padded rows 90..95 stay 0
    }
    __syncthreads();

    // ---- phase 2: dense normalized adjacency -------------------------------
    for (int i = tid; i < 96 * 96; i += 256) sA[i] = 0.0f;
    if (tid < 96) sDnv[tid] = (tid < ROIS) ? dinv[g * ROIS + tid] : 0.0f;
    __syncthreads();

    const int ebase = g * EPG;
    const int nbase = g * ROIS;
    for (int e = tid; e < EPG; e += 256) {
        int s = esrc[ebase + e] - nbase;
        int d = edst[ebase + e] - nbase;
        float nrm = sDnv[s] * ew[ebase + e] * sDnv[d];
        atomicAdd(&sA[d * 96 + s], nrm);         // ds_add_f32
    }
    __syncthreads();
    if (tid < ROIS) sA[tid * 96 + tid] += sDnv[tid] * sDnv[tid];
    __syncthreads();

    // ---- phase 3: OUT = A @ HW with f32 WMMA -------------------------------
    // 6 M-tiles x 4 N-tiles = 24 tiles; 8 waves take 3 tiles each.
    for (int t = wave; t < 24; t += 8) {
        const int tm = t >> 2, tn = t & 3;
        v8f c = {};
#pragma unroll 4
        for (int k = 0; k < 96; k += 4) {
            const int ka = k + 2 * hi;           // A: lanes0-15 K={k,k+1}, 16-31 K={k+2,k+3}
            v2f a, b;
            a.x = sA[(tm * 16 + l15) * 96 + ka];
            a.y = sA[(tm * 16 + l15) * 96 + ka + 1];
            b.x = sHW[ka * 64 + tn * 16 + l15];  // B rows striped across lanes
            b.y = sHW[(ka + 1) * 64 + tn * 16 + l15];
            c = __builtin_amdgcn_wmma_f32_16x16x4_f32(false, a, false, b,
                                                      (short)0, c, false, false);
        }
        const float bb = bias[tn * 16 + l15];
#pragma unroll
        for (int r = 0; r < 8; ++r) {            // D: VGPR r -> M=r (lo) / r+8 (hi)
            int row = tm * 16 + r + 8 * hi;
            if (row < ROIS) {
                float v = c[r] + bb;
                v = v > 0.0f ? v : 0.0f;
                img_out[(size_t)g * LIN1K + row * CATW + out_off + tn * 16 + l15] = v;
            }
        }
    }
}

// ------------------------------------------------------------ head GEMMs
// grid = 64 blocks (16-row tiles of B=1024), 256 thr = 8 waves.
// waves 0-3: lin1 N-tiles 0..3 ; waves 4-7: r1 N-tiles 0..3 (shared A tile).
// A tile (16x128 f32 = 8KB) staged via double-buffered async DMA to LDS.
__global__ __launch_bounds__(256)
void k_head1(const float* __restrict__ img_out,   // [1024][17280]
             const float* __restrict__ img_feat,  // [1024][288] zero-padded
             const float* __restrict__ lin1_W, const float* __restrict__ lin1_b,
             const float* __restrict__ r1_W,   const float* __restrict__ r1_b,
             float* __restrict__ h1, float* __restrict__ hr)
{
    __shared__ float sA[2][16 * 128];            // 2 x 8 KB chunks
    const int bt   = blockIdx.x;
    const int tid  = threadIdx.x;
    const int lane = tid & 31;
    const int wave = tid >> 5;
    const int l15  = lane & 15;
    const int hi   = lane >> 4;
    const int tn   = wave & 3;
    const bool is_r = (wave >= 4);
    const float* W = is_r ? r1_W : lin1_W;
    const int n = tn * 16 + l15;

    // stage one 16x128 chunk: 512 B128 segments, 2 per thread (in-order async)
    auto stage = [&](int buf, int kc) {
#pragma unroll
        for (int jj = 0; jj < 2; ++jj) {
            int j  = tid + jj * 256;             // [0,512)
            int r  = j >> 5;                     // row 0..15
            int cc = (j & 31) * 4;               // col 0..124
            async_ld_b128((unsigned)(uintptr_t)&sA[buf][j * 4],
                          &img_out[(size_t)(bt * 16 + r) * LIN1K + kc + cc]);
        }
    };

    stage(0, 0);
    int cur = 0;
    v8f c = {};
    for (int kc = 0; kc < LIN1K; kc += 128) {
        const bool has_next = (kc + 128 < LIN1K);
        if (has_next) {
            stage(cur ^ 1, kc + 128);            // overlap DMA with compute
            __builtin_prefetch(&W[(size_t)(kc + 128) * 64 + n], 0, 1);
            wait_async_le2();                    // chunk `cur` has landed
        } else {
            wait_async_le0();
        }
        __syncthreads();
        const float* sAc = sA[cur];
#pragma unroll 4
        for (int k = 0; k < 128; k += 4) {
            const int ka = k + 2 * hi;
            v2f a, b;
            a.x = sAc[l15 * 128 + ka];
            a.y = sAc[l15 * 128 + ka + 1];
            b.x = W[(size_t)(kc + ka) * 64 + n];
            b.y = W[(size_t)(kc + ka + 1) * 64 + n];
            c = __builtin_amdgcn_wmma_f32_16x16x4_f32(false, a, false, b,
                                                      (short)0, c, false, false);
        }
        __syncthreads();                         // buf reuse safe next iter
        cur ^= 1;
    }
    if (is_r) {
        // regression tail: 270 gated raw features (padded to 288 with zeros)
        for (int k = 0; k < FEATP; k += 4) {
            const int ka = k + 2 * hi;
            v2f a, b;
            a.x = img_feat[(size_t)(bt * 16 + l15) * FEATP + ka];
            a.y = img_feat[(size_t)(bt * 16 + l15) * FEATP + ka + 1];
            b.x = (ka     < FEATV) ? W[(size_t)(LIN1K + ka)     * 64 + n] : 0.0f;
            b.y = (ka + 1 < FEATV) ? W[(size_t)(LIN1K + ka + 1) * 64 + n] : 0.0f;
            c = __builtin_amdgcn_wmma_f32_16x16x4_f32(false, a, false, b,
                                                      (short)0, c, false, false);
        }
    }
    const float bb = is_r ? r1_b[n] : lin1_b[n];
    float* dst = is_r ? hr : h1;
#pragma unroll
    for (int r = 0; r < 8; ++r) {
        int row = bt * 16 + r + 8 * hi;
        float v = c[r] + bb;
        dst[(size_t)row * 64 + n] = v > 0.0f ? v : 0.0f;
    }
}

// tiny output GEMMs: [1024,64]@[64,2] and [1024,64]@[64,4] -> out[1024][6]
__global__ void k_head2(const float* __restrict__ h1, const float* __restrict__ hr,
                        const float* __restrict__ lin2_W, const float* __restrict__ lin2_b,
                        const float* __restrict__ r2_W,  const float* __restrict__ r2_b,
                        float* __restrict__ out)
{
    int b = blockIdx.x * blockDim.x + threadIdx.x;
    if (b >= BGR) return;
    float acc0 = lin2_b[0], acc1 = lin2_b[1];
    float acc2 = r2_b[0], acc3 = r2_b[1], acc4 = r2_b[2], acc5 = r2_b[3];
#pragma unroll 4
    for (int k = 0; k < 64; ++k) {
        float hv = h1[(size_t)b * 64 + k];
        acc0 = fmaf(hv, lin2_W[k * 2 + 0], acc0);
        acc1 = fmaf(hv, lin2_W[k * 2 + 1], acc1);
        float rv = hr[(size_t)b * 64 + k];
        acc2 = fmaf(rv, r2_W[k * 4 + 0], acc2);
        acc3 = fmaf(rv, r2_W[k * 4 + 1], acc3);
        acc4 = fmaf(rv, r2_W[k * 4 + 2], acc4);
        acc5 = fmaf(rv, r2_W[k * 4 + 3], acc5);
    }
    out[(size_t)b * 6 + 0] = acc0;
    out[(size_t)b * 6 + 1] = acc1;
    out[(size_t)b * 6 + 2] = acc2;
    out[(size_t)b * 6 + 3] = acc3;
    out[(size_t)b * 6 + 4] = acc4;
    out[(size_t)b * 6 + 5] = acc5;
}

// --------------------------------------------------------------------------
extern "C" void kernel_launch(void* const* d_in, const int* in_sizes, int n_in,
                              void* d_out, int out_size, void* d_ws, size_t ws_size,
                              hipStream_t stream) {
    const float* x      = (const float*)d_in[0];
    const int*   esrc   = (const int*)  d_in[1];
    const int*   edst   = (const int*)  d_in[2];
    const float* ew     = (const float*)d_in[3];
    const float* W1     = (const float*)d_in[4];
    const float* b1     = (const float*)d_in[5];
    const float* W2     = (const float*)d_in[6];
    const float* b2     = (const float*)d_in[7];
    const float* W3     = (const float*)d_in[8];
    const float* b3     = (const float*)d_in[9];
    const float* prob   = (const float*)d_in[10];
    const float* lin1_W = (const float*)d_in[11];
    const float* lin1_b = (const float*)d_in[12];
    const float* lin2_W = (const float*)d_in[13];
    const float* lin2_b = (const float*)d_in[14];
    const float* r1_W   = (const float*)d_in[15];
    const float* r1_b   = (const float*)d_in[16];
    const float* r2_W   = (const float*)d_in[17];
    const float* r2_b   = (const float*)d_in[18];
    float* out = (float*)d_out;

    // workspace layout (floats); total ~73.2 MB
    float* deg      = (float*)d_ws;
    float* dinv     = deg + NNODES;
    float* img_out  = dinv + NNODES;                 // [1024][17280]
    float* img_feat = img_out + (size_t)BGR * LIN1K; // [1024][288]
    float* h1       = img_feat + (size_t)BGR * FEATP;
    float* hr       = h1 + (size_t)BGR * 64;

    // 1) degree + normalization
    k_zero<<<(NNODES + 255) / 256, 256, 0, stream>>>(deg, NNODES);
    k_deg <<<NEDGE / 256, 256, 0, stream>>>(edst, ew, deg);
    k_dinv<<<(NNODES + 255) / 256, 256, 0, stream>>>(deg, dinv);

    // 2) three GCN layers (one block per graph)
    k_gcn<3> <<<BGR, 256, 0, stream>>>(x,       ROIS * 3, 3,    0,
                                       W1, b1, esrc, edst, ew, dinv, img_out, 0);
    k_gcn<64><<<BGR, 256, 0, stream>>>(img_out, LIN1K,    CATW, 0,
                                       W2, b2, esrc, edst, ew, dinv, img_out, 64);
    k_gcn<64><<<BGR, 256, 0, stream>>>(img_out, LIN1K,    CATW, 64,
                                       W3, b3, esrc, edst, ew, dinv, img_out, 128);

    // 3) gated raw features for regression head (zero-padded to 288 cols)
    k_gate<<<(BGR * FEATP + 255) / 256, 256, 0, stream>>>(x, prob, img_feat);

    // 4) dual wide head GEMM (WMMA + async LDS staging), tiny output GEMMs
    k_head1<<<BGR / 16, 256, 0, stream>>>(img_out, img_feat,
                                          lin1_W, lin1_b, r1_W, r1_b, h1, hr);
    k_head2<<<BGR / 256, 256, 0, stream>>>(h1, hr, lin2_W, lin2_b, r2_W, r2_b, out);
}